// Model_7868380086954
// MI455X (gfx1250) — compile-verified
//
#include <hip/hip_runtime.h>
#include <hip/hip_bf16.h>
#include <stdint.h>

// Problem constants from the reference
#define B_ 8
#define T_ 512
#define H_ 8
#define D_ 128
#define S_ 8192

#define COPY_GRID 4096u   // blocks per tensor in the copy kernel

typedef float v4f __attribute__((ext_vector_type(4)));
typedef float v2f __attribute__((ext_vector_type(2)));

// ---------------------------------------------------------------------------
// Kernel 1: bulk cache -> out copy over the CDNA5 async global<->LDS DMA path.
// One launch covers both tensors: blockIdx.y = 0 -> cache_k, 1 -> cache_v.
//
// Each thread owns 4 private 16B LDS staging slots. Per block-iteration the
// block moves 256 threads * 4 * 16B = 16 KB. ASYNCcnt is per-wave and each
// wave only touches its own lanes' LDS slots, so no barriers are needed:
//   issue 4x GLOBAL_LOAD_ASYNC_TO_LDS_B128   (ASYNCcnt += 4)
//   s_wait_asynccnt 0                        (LDS now holds the data)
//   issue 4x GLOBAL_STORE_ASYNC_FROM_LDS_B128
//   s_wait_asynccnt 0                        (LDS reusable next iteration)
// Non-temporal hints on both sides: data is streamed exactly once, keep the
// 192MB L2 clean and let store lines rinse to HBM.
// iters is precomputed on the host (no 64-bit division in the preamble);
// n4 divides evenly, so every async op runs with EXEC all-1s (no tail).
// ---------------------------------------------------------------------------
__global__ __launch_bounds__(256) void bulk_copy_async(const v4f* __restrict__ srcK,
                                                       v4f* __restrict__ dstK,
                                                       const v4f* __restrict__ srcV,
                                                       v4f* __restrict__ dstV,
                                                       unsigned iters) {
  __shared__ alignas(16) v4f smem[256 * 4];
  const unsigned tid = threadIdx.x;
  // Low 32 bits of the flat shared-aperture address == LDS byte offset.
  const unsigned lds0 = (unsigned)(uintptr_t)(&smem[tid]);

  const v4f* __restrict__ src = (blockIdx.y == 0) ? srcK : srcV;
  v4f* __restrict__ dst       = (blockIdx.y == 0) ? dstK : dstV;

  const unsigned long long chunk = 256ull * 4ull;        // float4 per block-iteration
  const unsigned long long step  = (unsigned long long)COPY_GRID * chunk;

  const v4f* s = src + (unsigned long long)blockIdx.x * chunk + tid;
  v4f*       d = dst + (unsigned long long)blockIdx.x * chunk + tid;

  for (unsigned it = 0; it < iters; ++it) {
#pragma unroll
    for (int k = 0; k < 4; ++k) {
      const unsigned long long gsrc = (unsigned long long)(uintptr_t)(s + k * 256);
      const unsigned lds = lds0 + (unsigned)(k * 256 * 16);
      asm volatile("global_load_async_to_lds_b128 %0, %1, off th:TH_LOAD_NT"
                   :: "v"(lds), "v"(gsrc)
                   : "memory");
    }
    asm volatile("s_wait_asynccnt 0x0" ::: "memory");
#pragma unroll
    for (int k = 0; k < 4; ++k) {
      const unsigned long long gdst = (unsigned long long)(uintptr_t)(d + k * 256);
      const unsigned lds = lds0 + (unsigned)(k * 256 * 16);
      asm volatile("global_store_async_from_lds_b128 %0, %1, off th:TH_STORE_NT"
                   :: "v"(gdst), "v"(lds)
                   : "memory");
    }
    asm volatile("s_wait_asynccnt 0x0" ::: "memory");
    s += step;
    d += step;
  }
}

// ---------------------------------------------------------------------------
// Kernel 2: RoPE rotate k_new and scatter (k,v) rows into the output caches.
//
// One wave32 per (b,t,h) row: 32 lanes * float4 = 128 floats = one D row.
// Lane l handles elements [4l..4l+3] = RoPE pairs 2l and 2l+1, so cos/sin
// arrive as one aligned v2f per lane. Stores are nontemporal B128 (rows are
// written once and never re-read -> keep L2 for the small cos/sin tables).
// ---------------------------------------------------------------------------
__global__ __launch_bounds__(256) void rope_scatter(const float* __restrict__ k_new,
                                                    const float* __restrict__ v_new,
                                                    const float* __restrict__ cosp,
                                                    const float* __restrict__ sinp,
                                                    const long long* __restrict__ positions,
                                                    float* __restrict__ out_k,
                                                    float* __restrict__ out_v) {
  const int lane = threadIdx.x & 31;
  const int wave = threadIdx.x >> 5;
  const int row  = blockIdx.x * 8 + wave;        // row in [0, B*T*H)
  const int h  = row & (H_ - 1);
  const int bt = row >> 3;                       // b*T + t   (H_ == 8)
  const int b  = bt / T_;

  const long long pos = positions[bt];

  const v4f* kin = (const v4f*)(k_new + (size_t)row * D_);
  const v4f* vin = (const v4f*)(v_new + (size_t)row * D_);
  const v4f kv = __builtin_nontemporal_load(kin + lane);
  const v4f vv = __builtin_nontemporal_load(vin + lane);

  const v2f c = *(const v2f*)(cosp + (size_t)pos * (D_ / 2) + lane * 2);
  const v2f s = *(const v2f*)(sinp + (size_t)pos * (D_ / 2) + lane * 2);

  v4f r;
  r.x = kv.x * c.x - kv.y * s.x;   // rot_even(pair 2*lane)
  r.y = kv.x * s.x + kv.y * c.x;   // rot_odd (pair 2*lane)
  r.z = kv.z * c.y - kv.w * s.y;   // rot_even(pair 2*lane+1)
  r.w = kv.z * s.y + kv.w * c.y;   // rot_odd (pair 2*lane+1)

  const size_t dst = (((size_t)b * S_ + (size_t)pos) * H_ + (size_t)h) * D_;
  __builtin_nontemporal_store(r,  (v4f*)(out_k + dst) + lane);
  __builtin_nontemporal_store(vv, (v4f*)(out_v + dst) + lane);
}

extern "C" void kernel_launch(void* const* d_in, const int* in_sizes, int n_in,
                              void* d_out, int out_size, void* d_ws, size_t ws_size,
                              hipStream_t stream) {
  const float*     k_new     = (const float*)d_in[0];
  const float*     v_new     = (const float*)d_in[1];
  const float*     cosp      = (const float*)d_in[2];
  const float*     sinp      = (const float*)d_in[3];
  const float*     cache_k   = (const float*)d_in[4];
  const float*     cache_v   = (const float*)d_in[5];
  const long long* positions = (const long long*)d_in[6];

  float* out_k = (float*)d_out;
  float* out_v = out_k + (size_t)B_ * S_ * H_ * D_;

  const unsigned long long n4 = (unsigned long long)B_ * S_ * H_ * D_ / 4; // 16,777,216
  const unsigned iters = (unsigned)(n4 / ((unsigned long long)COPY_GRID * 1024ull)); // == 4

  dim3 block(256);
  // (4096 blocks x 2 tensors) * 8 wave32 = 64K waves of in-flight async DMA;
  // each block runs `iters` counted chunks of 16KB.
  bulk_copy_async<<<dim3(COPY_GRID, 2), block, 0, stream>>>(
      (const v4f*)cache_k, (v4f*)out_k, (const v4f*)cache_v, (v4f*)out_v, iters);

  // B*T*H = 32768 rows, 8 rows (waves) per block
  rope_scatter<<<dim3((B_ * T_ * H_) / 8), block, 0, stream>>>(
      k_new, v_new, cosp, sinp, positions, out_k, out_v);
}